// larange_26843545600036
// MI455X (gfx1250) — compile-verified
//
#include <hip/hip_runtime.h>

typedef __attribute__((ext_vector_type(2))) float v2f;
typedef __attribute__((ext_vector_type(8))) float v8f;

#define C_DIM 128
#define S_DIM 31
#define HW    704      // 32*22
#define SHW   21824    // 31*704
#define NPIX  87296    // 4*31*704
#define NTILES 5456    // NPIX/16 (exact)
#define WPB   8        // waves per block (kernel 1)
#define NS    120      // 4*(31-1)
#define SORTN 1024
#define SPLIT 2
#define IPB   352      // 704/SPLIT, threads per block (kernel 2) = 11 waves

// Kernel 1: Y1[p] = dot(x[p,:],w1)+b1 ; Y2[p] = dot(x[p,:],w2)+b2 for all
// n*s*h*w pixels, via V_WMMA_F32_16X16X4_F32. A holds the 2x128 weight matrix
// (rows M=0,1 of 16; zero rows produced by a 0/1 lane mask instead of
// branches), B holds 16 pixels x 4 channels, accumulate K=128 in 32 WMMA
// steps. D row 0 -> Y1, row 1 -> Y2 (lanes 0-15, N = pixel).
__global__ __launch_bounds__(256) void conv1x1_wmma(
    const float* __restrict__ x,  const float* __restrict__ w1,
    const float* __restrict__ b1, const float* __restrict__ w2,
    const float* __restrict__ b2, float* __restrict__ Y1,
    float* __restrict__ Y2)
{
    const int lane = threadIdx.x & 31;
    const int wave = threadIdx.x >> 5;
    const int tile = blockIdx.x * WPB + wave;
    if (tile >= NTILES) return;               // never taken (exact grid), whole-wave uniform
    const int pixBase = tile * 16;
    const int col  = lane & 15;               // M (weight row) for A, N (pixel) for B
    const int half = lane >> 4;               // K-slot group: 0 -> K0,K1 ; 1 -> K2,K3
    const int p     = pixBase + col;
    const int n_idx = p / SHW;
    const int rem   = p - n_idx * SHW;
    const float* xb = x + (size_t)n_idx * C_DIM * SHW + rem;

    // Branch-free A-fragment source: pointer selected once, zero rows via mask.
    const float* wptr = (col == 1) ? w2 : w1;
    const float  msk  = (col < 2) ? 1.0f : 0.0f;
    const float  b1v  = b1[0];
    const float  b2v  = b2[0];

    v8f acc = {};
    #pragma unroll 8
    for (int k = 0; k < C_DIM; k += 4) {
        const int koff = k + 2 * half;        // even -> 8B aligned
        const v2f wv = *(const v2f*)(wptr + koff);   // broadcast-friendly b64 load
        v2f a; a.x = wv.x * msk; a.y = wv.y * msk;
        // B: 4x16 pixel-channel tile; coalesced: 16 lanes read 16 consecutive pixels
        v2f b;
        b.x = xb[(size_t)koff * SHW];
        b.y = xb[(size_t)(koff + 1) * SHW];
        acc = __builtin_amdgcn_wmma_f32_16x16x4_f32(
            false, a, false, b, (short)0, acc, false, false);
    }
    // D layout: VGPR r, lanes 0-15 -> M=r, N=lane. M=0 row = Y1, M=1 row = Y2.
    if (lane < 16) {
        Y1[pixBase + lane] = acc[0] + b1v;
        Y2[pixBase + lane] = acc[1] + b2v;
    }
}

// Kernel 2: per row r=(n,f): sort X1 (frame f+1 of Y2) descending in LDS.
// score[i,j] = X0[i]*X1[j] is rank-1, and softmax is monotonic, so
// top_k(softmax) = softmax evaluated at top_k(score):
//   a = X0[i]; m = a>=0 ? a*max(X1) : a*min(X1); denom = sum_j exp(a*X1[j]-m)
//   k-th value (descending) = exp(a*sk - m)/denom,
//   sk = a>=0 ? k-th largest : k-th smallest of X1.
// a==0 degenerates to uniform 1/704 automatically.
__global__ __launch_bounds__(IPB) void attn_topk(
    const float* __restrict__ Y1, const float* __restrict__ Y2,
    float* __restrict__ out)
{
    __shared__ float s1[SORTN];
    const int r     = blockIdx.x / SPLIT;
    const int part  = blockIdx.x % SPLIT;
    const int n_idx = r / 30;
    const int f     = r % 30;
    const float* X0 = Y1 + (size_t)(n_idx * S_DIM + f) * HW;
    const float* X1 = Y2 + (size_t)(n_idx * S_DIM + f + 1) * HW;
    const int tid = threadIdx.x;

    for (int idx = tid; idx < SORTN; idx += IPB)
        s1[idx] = (idx < HW) ? X1[idx] : -__builtin_inff();
    __syncthreads();

    // Bitonic sort, overall descending; -inf pads land at the tail.
    for (unsigned k = 2; k <= SORTN; k <<= 1) {
        for (unsigned j = k >> 1; j > 0; j >>= 1) {
            for (unsigned i = tid; i < SORTN; i += IPB) {
                const unsigned ixj = i ^ j;
                if (ixj > i) {
                    const float va = s1[i], vb = s1[ixj];
                    const bool desc = ((i & k) == 0);
                    if (desc ? (va < vb) : (va > vb)) { s1[i] = vb; s1[ixj] = va; }
                }
            }
            __syncthreads();
        }
    }

    const int   i = part * IPB + tid;          // one thread per output column i
    const float a = X0[i];
    const float m = (a >= 0.f) ? a * s1[0] : a * s1[HW - 1];
    float denom = 0.f;
    #pragma unroll 4
    for (int j = 0; j < HW; ++j)               // uniform j -> LDS broadcast reads
        denom += __expf(a * s1[j] - m);
    const float inv = 1.f / denom;

    #pragma unroll
    for (int k2 = 0; k2 < 16; ++k2) {
        const float sk  = (a >= 0.f) ? s1[k2] : s1[HW - 1 - k2];
        const float val = __expf(a * sk - m) * inv;
        // out shape (n, 16, s-1, h, w)
        out[(((size_t)(n_idx * 16 + k2)) * 30 + f) * HW + i] = val;
    }
}

extern "C" void kernel_launch(void* const* d_in, const int* in_sizes, int n_in,
                              void* d_out, int out_size, void* d_ws, size_t ws_size,
                              hipStream_t stream) {
    (void)in_sizes; (void)n_in; (void)out_size; (void)ws_size;
    const float* x  = (const float*)d_in[0];
    const float* w1 = (const float*)d_in[1];
    const float* b1 = (const float*)d_in[2];
    const float* w2 = (const float*)d_in[3];
    const float* b2 = (const float*)d_in[4];
    float* Y1 = (float*)d_ws;          // NPIX floats
    float* Y2 = Y1 + NPIX;             // NPIX floats (total 698 KB scratch)
    float* out = (float*)d_out;

    conv1x1_wmma<<<NTILES / WPB, 256, 0, stream>>>(x, w1, b1, w2, b2, Y1, Y2);
    attn_topk<<<NS * SPLIT, IPB, 0, stream>>>(Y1, Y2, out);
}